// TransformerBlock_28243704939177
// MI455X (gfx1250) — compile-verified
//
#include <hip/hip_runtime.h>
#include <math.h>

// ---------------------------------------------------------------------------
// MLA transformer block for MI455X (gfx1250), fp32 end-to-end.
// All GEMMs run on V_WMMA_F32_16X16X4_F32 (wave32 matrix pipe).
// Tiles: BM=128, BN=64, BK=16; 8 waves/block; each wave owns a 32x32 C tile
// (4 f32 WMMA accumulator chains). Global tile loads are b128 and double-
// buffered through LDS (ping-pong): next tile's global loads are issued
// before the current tile's WMMA chain, so global latency hides behind the
// matrix pipe and only one barrier per K-tile is needed. LDS layout is
// [k][m] (+4 pad): bank-conflict-free for both the scattered b32 stores and
// the fragment reads. All shapes are multiples of the tile (no bounds checks).
// ---------------------------------------------------------------------------

typedef __attribute__((ext_vector_type(2))) float v2f;
typedef __attribute__((ext_vector_type(4))) float v4f;
typedef __attribute__((ext_vector_type(8))) float v8f;

#define B_  2
#define S_  1024
#define D_  1024
#define H_  16
#define QL_ 512
#define KVL_ 512
#define NOPE_ 128
#define ROPE_ 64
#define QKH_ 192
#define VH_ 128
#define E_  8
#define DFF_ 2048
#define R_  (B_ * S_)          // 2048 token rows

// Epilogue selectors
#define EP_NONE         0
#define EP_BIAS         1
#define EP_BIAS_RES     2
#define EP_SILU         3
#define EP_BIAS_MUL     4
#define EP_ROWSCALE_ACC 5
#define EP_SCALE_MASK   6

// batch-offset descriptor: offset(z) = (z/div)*s1 + (z%div)*s2
struct BOff { long s1; long s2; int div; };
__host__ __device__ inline long boff(const BOff& b, int z) {
    return (long)(z / b.div) * b.s1 + (long)(z % b.div) * b.s2;
}

// ---------------------------------------------------------------------------
// Generic tiled WMMA GEMM:  C = epilogue(A[M,K] x Bmat[K,N])
//   TRANSB=1: Bmat[k][n] = W[n*ldw + k]
//   TRANSB=0: Bmat[k][n] = W[k*ldw + n]
// Requires: M % 128 == 0, N % 64 == 0, K % 16 == 0, 16B-aligned pointers.
// ---------------------------------------------------------------------------
template<int EP, int TRANSB>
__global__ __launch_bounds__(256)
void gemm_wmma_kernel(const float* __restrict__ A, int lda, BOff ba,
                      const float* __restrict__ W, int ldw, BOff bw,
                      float* __restrict__ C, int ldc, BOff bc,
                      const float* __restrict__ bias, BOff bbias,
                      const float* __restrict__ extra, int lde, BOff bext,
                      const float* __restrict__ rowscale, int rs_stride, BOff brs,
                      float scale,
                      int M, int N, int Kd)
{
    constexpr int BM = 128, BN = 64, BK = 16;
    __shared__ __align__(16) float As[2][BK][BM + 4];
    __shared__ __align__(16) float Bs[2][BK][BN + 4];

    const int z = blockIdx.z;
    A += boff(ba, z);
    W += boff(bw, z);
    C += boff(bc, z);
    if (bias)     bias     += boff(bbias, z);
    if (extra)    extra    += boff(bext, z);
    if (rowscale) rowscale += boff(brs, z);

    const int tid  = threadIdx.x;
    const int wid  = tid >> 5;
    const int lane = tid & 31;
    const int wm   = wid & 3;      // 0..3 : 32-row strip
    const int wn   = wid >> 2;     // 0..1 : 32-col strip
    const int blockM = blockIdx.y * BM;
    const int blockN = blockIdx.x * BN;

    const bool hi = lane >= 16;
    const int  lm = lane & 15;

    v8f acc00 = {0,0,0,0,0,0,0,0};   // rows +0..15, cols +0..15
    v8f acc01 = {0,0,0,0,0,0,0,0};   // rows +0..15, cols +16..31
    v8f acc10 = {0,0,0,0,0,0,0,0};   // rows +16..31, cols +0..15
    v8f acc11 = {0,0,0,0,0,0,0,0};   // rows +16..31, cols +16..31

    // A-tile load indices: 2 x b128 per thread (128 rows x 16 k)
    const int am0 = tid >> 2;            // 0..63
    const int ak  = (tid & 3) * 4;       // 0,4,8,12
    // B-tile load indices
    const int bn_t = tid >> 2;           // TRANSB=1: n index 0..63
    const int bk_t = (tid & 3) * 4;      // TRANSB=1: k group
    const int bk_n = tid >> 4;           // TRANSB=0: k index 0..15
    const int bn_n = (tid & 15) * 4;     // TRANSB=0: n group

    const float* Arow0 = A + (long)(blockM + am0) * lda;
    const float* Arow1 = A + (long)(blockM + am0 + 64) * lda;

    v4f aV0, aV1, bV;
    // staged global load of tile at k0
    auto gload = [&](int k0) {
        aV0 = *(const v4f*)(Arow0 + k0 + ak);
        aV1 = *(const v4f*)(Arow1 + k0 + ak);
        if (TRANSB)
            bV = *(const v4f*)(W + (long)(blockN + bn_t) * ldw + k0 + bk_t);
        else
            bV = *(const v4f*)(W + (long)(k0 + bk_n) * ldw + blockN + bn_n);
    };
    // commit staged tile to LDS buffer `buf`
    auto sstore = [&](int buf) {
        As[buf][ak + 0][am0] = aV0.x;
        As[buf][ak + 1][am0] = aV0.y;
        As[buf][ak + 2][am0] = aV0.z;
        As[buf][ak + 3][am0] = aV0.w;
        As[buf][ak + 0][am0 + 64] = aV1.x;
        As[buf][ak + 1][am0 + 64] = aV1.y;
        As[buf][ak + 2][am0 + 64] = aV1.z;
        As[buf][ak + 3][am0 + 64] = aV1.w;
        if (TRANSB) {
            Bs[buf][bk_t + 0][bn_t] = bV.x;
            Bs[buf][bk_t + 1][bn_t] = bV.y;
            Bs[buf][bk_t + 2][bn_t] = bV.z;
            Bs[buf][bk_t + 3][bn_t] = bV.w;
        } else {
            *(v4f*)&Bs[buf][bk_n][bn_n] = bV;
        }
    };

    const int nTiles = Kd >> 4;
    gload(0);
    sstore(0);
    __syncthreads();

    for (int kt = 0; kt < nTiles; ++kt) {
        const int cur = kt & 1;
        const bool more = (kt + 1) < nTiles;
        if (more) gload((kt + 1) << 4);   // prefetch next tile into registers

        #pragma unroll
        for (int k4 = 0; k4 < 4; ++k4) {
            // A frag (16x4): lanes 0-15 hold K pair {0,1}, lanes 16-31 hold {2,3}
            int kb = k4 * 4 + (hi ? 2 : 0);
            v2f a0, a1, b0, b1;
            a0.x = As[cur][kb][wm * 32 + lm];
            a0.y = As[cur][kb + 1][wm * 32 + lm];
            a1.x = As[cur][kb][wm * 32 + 16 + lm];
            a1.y = As[cur][kb + 1][wm * 32 + 16 + lm];
            b0.x = Bs[cur][kb][wn * 32 + lm];
            b0.y = Bs[cur][kb + 1][wn * 32 + lm];
            b1.x = Bs[cur][kb][wn * 32 + 16 + lm];
            b1.y = Bs[cur][kb + 1][wn * 32 + 16 + lm];
            acc00 = __builtin_amdgcn_wmma_f32_16x16x4_f32(
                false, a0, false, b0, (short)0, acc00, false, false);
            acc01 = __builtin_amdgcn_wmma_f32_16x16x4_f32(
                false, a0, false, b1, (short)0, acc01, false, false);
            acc10 = __builtin_amdgcn_wmma_f32_16x16x4_f32(
                false, a1, false, b0, (short)0, acc10, false, false);
            acc11 = __builtin_amdgcn_wmma_f32_16x16x4_f32(
                false, a1, false, b1, (short)0, acc11, false, false);
        }

        if (more) {
            sstore(cur ^ 1);   // s_wait_loadcnt lands here, behind the WMMAs
            __syncthreads();
        }
    }

    // epilogue: C/D layout — VGPR r: lanes 0-15 -> row r, lanes 16-31 -> row r+8
    #pragma unroll
    for (int ms = 0; ms < 2; ++ms) {
        #pragma unroll
        for (int r = 0; r < 8; ++r) {
            int row = blockM + wm * 32 + ms * 16 + r + (hi ? 8 : 0);
            #pragma unroll
            for (int nt = 0; nt < 2; ++nt) {
                int col = blockN + wn * 32 + nt * 16 + lm;
                float a;
                if (ms == 0) a = (nt == 0) ? acc00[r] : acc01[r];
                else         a = (nt == 0) ? acc10[r] : acc11[r];
                long ci = (long)row * ldc + col;
                if (EP == EP_NONE) {
                    C[ci] = a;
                } else if (EP == EP_BIAS) {
                    C[ci] = a + bias[col];
                } else if (EP == EP_BIAS_RES) {
                    C[ci] = a + bias[col] + extra[(long)row * lde + col];
                } else if (EP == EP_SILU) {
                    C[ci] = a / (1.f + __expf(-a));
                } else if (EP == EP_BIAS_MUL) {
                    C[ci] = (a + bias[col]) * extra[(long)row * lde + col];
                } else if (EP == EP_ROWSCALE_ACC) {
                    C[ci] += rowscale[(long)row * rs_stride] * a;
                } else if (EP == EP_SCALE_MASK) {
                    C[ci] = a * scale + extra[(long)row * lde + col];
                }
            }
        }
    }
}

// ---------------------------------------------------------------------------
// RMSNorm: one block per row.  out[row, 0..Dn) = x * rsqrt(mean(x^2)+eps) * w
// ---------------------------------------------------------------------------
__global__ __launch_bounds__(256)
void rms_kernel(const float* __restrict__ in, long ldi,
                const float* __restrict__ w,
                float* __restrict__ out, long ldo, int Dn)
{
    __shared__ float red[256];
    long row = blockIdx.x;
    const float* x = in + row * ldi;
    float s = 0.f;
    for (int c = threadIdx.x; c < Dn; c += 256) { float v = x[c]; s += v * v; }
    red[threadIdx.x] = s;
    __syncthreads();
    for (int off = 128; off > 0; off >>= 1) {
        if (threadIdx.x < off) red[threadIdx.x] += red[threadIdx.x + off];
        __syncthreads();
    }
    float r = rsqrtf(red[0] / (float)Dn + 1e-6f);
    for (int c = threadIdx.x; c < Dn; c += 256) out[row * ldo + c] = x[c] * r * w[c];
}

// ---------------------------------------------------------------------------
// RoPE on q_pe (q cols [h*192+128, h*192+192)) -> q_ext cols [512,576)
// ---------------------------------------------------------------------------
__global__ void rope_q_kernel(const float* __restrict__ q,
                              const float* __restrict__ fcos,
                              const float* __restrict__ fsin,
                              float* __restrict__ q_ext)
{
    int idx = blockIdx.x * blockDim.x + threadIdx.x;
    if (idx >= R_ * H_ * (ROPE_ / 2)) return;
    int j = idx & 31;
    int h = (idx >> 5) & 15;
    int row = idx >> 9;
    int s = row & (S_ - 1);
    const float* base = q + (long)row * (H_ * QKH_) + h * QKH_ + NOPE_;
    float xr = base[2 * j], xi = base[2 * j + 1];
    float c = fcos[s * 32 + j], sn = fsin[s * 32 + j];
    float* ob = q_ext + (long)row * (H_ * 576) + h * 576 + 512;
    ob[2 * j]     = xr * c - xi * sn;
    ob[2 * j + 1] = xr * sn + xi * c;
}

// RoPE on k_pe (kv cols [512,576)) -> k_ext cols [512,576)
__global__ void rope_k_kernel(const float* __restrict__ kv,
                              const float* __restrict__ fcos,
                              const float* __restrict__ fsin,
                              float* __restrict__ k_ext)
{
    int idx = blockIdx.x * blockDim.x + threadIdx.x;
    if (idx >= R_ * (ROPE_ / 2)) return;
    int j = idx & 31;
    int row = idx >> 5;
    int s = row & (S_ - 1);
    const float* base = kv + (long)row * 576 + 512;
    float xr = base[2 * j], xi = base[2 * j + 1];
    float c = fcos[s * 32 + j], sn = fsin[s * 32 + j];
    float* ob = k_ext + (long)row * 576 + 512;
    ob[2 * j]     = xr * c - xi * sn;
    ob[2 * j + 1] = xr * sn + xi * c;
}

// ---------------------------------------------------------------------------
// Row softmax (in place), one block per row.
// ---------------------------------------------------------------------------
__global__ __launch_bounds__(256)
void softmax_kernel(float* __restrict__ p, int cols)
{
    __shared__ float red[256];
    long row = blockIdx.x;
    float* x = p + row * (long)cols;
    float m = -3.4e38f;
    for (int c = threadIdx.x; c < cols; c += 256) m = fmaxf(m, x[c]);
    red[threadIdx.x] = m;
    __syncthreads();
    for (int off = 128; off > 0; off >>= 1) {
        if (threadIdx.x < off)
            red[threadIdx.x] = fmaxf(red[threadIdx.x], red[threadIdx.x + off]);
        __syncthreads();
    }
    m = red[0];
    __syncthreads();
    float s = 0.f;
    for (int c = threadIdx.x; c < cols; c += 256) {
        float e = __expf(x[c] - m);
        x[c] = e;
        s += e;
    }
    red[threadIdx.x] = s;
    __syncthreads();
    for (int off = 128; off > 0; off >>= 1) {
        if (threadIdx.x < off) red[threadIdx.x] += red[threadIdx.x + off];
        __syncthreads();
    }
    float inv = 1.f / red[0];
    for (int c = threadIdx.x; c < cols; c += 256) x[c] *= inv;
}

// ---------------------------------------------------------------------------
// Gate: logits = h2 @ gate_w.T (8 experts), top-2 softmax -> combine[row][8]
// one wave per token row.
// ---------------------------------------------------------------------------
__global__ __launch_bounds__(256)
void gate_kernel(const float* __restrict__ h2,
                 const float* __restrict__ gw,
                 float* __restrict__ combine)
{
    int wid = threadIdx.x >> 5;
    int lane = threadIdx.x & 31;
    long row = (long)blockIdx.x * 8 + wid;
    if (row >= R_) return;
    const float* x = h2 + row * D_;
    float acc[E_];
    #pragma unroll
    for (int e = 0; e < E_; ++e) acc[e] = 0.f;
    for (int d = lane; d < D_; d += 32) {
        float xv = x[d];
        #pragma unroll
        for (int e = 0; e < E_; ++e) acc[e] += xv * gw[e * D_ + d];
    }
    #pragma unroll
    for (int e = 0; e < E_; ++e)
        for (int off = 16; off > 0; off >>= 1)
            acc[e] += __shfl_xor(acc[e], off, 32);
    if (lane == 0) {
        int i1 = 0; float v1 = acc[0];
        for (int e = 1; e < E_; ++e) if (acc[e] > v1) { v1 = acc[e]; i1 = e; }
        int i2 = -1; float v2 = -3.4e38f;
        for (int e = 0; e < E_; ++e)
            if (e != i1 && acc[e] > v2) { v2 = acc[e]; i2 = e; }
        float e1 = __expf(v1 - v1), e2 = __expf(v2 - v1);
        float inv = 1.f / (e1 + e2);
        float* out = combine + row * E_;
        #pragma unroll
        for (int e = 0; e < E_; ++e) out[e] = 0.f;
        out[i1] = e1 * inv;
        out[i2] = e2 * inv;
    }
}

// ---------------------------------------------------------------------------
// Host launch
// ---------------------------------------------------------------------------
static const BOff BZ = {0, 0, 1};   // no batch offset

extern "C" void kernel_launch(void* const* d_in, const int* in_sizes, int n_in,
                              void* d_out, int out_size, void* d_ws, size_t ws_size,
                              hipStream_t stream) {
    (void)in_sizes; (void)n_in; (void)out_size; (void)ws_size;

    const float* x        = (const float*)d_in[0];
    const float* fcos     = (const float*)d_in[1];
    const float* fsin     = (const float*)d_in[2];
    const float* mask     = (const float*)d_in[3];
    const float* norm1_w  = (const float*)d_in[4];
    const float* wq_a_w   = (const float*)d_in[5];
    const float* wq_a_b   = (const float*)d_in[6];
    const float* q_norm_w = (const float*)d_in[7];
    const float* wq_b_w   = (const float*)d_in[8];
    const float* wq_b_b   = (const float*)d_in[9];
    const float* wkv_a_w  = (const float*)d_in[10];
    const float* wkv_a_b  = (const float*)d_in[11];
    const float* kv_norm_w= (const float*)d_in[12];
    const float* wkv_b_w  = (const float*)d_in[13];
    const float* wo_w     = (const float*)d_in[14];
    const float* wo_b     = (const float*)d_in[15];
    const float* norm2_w  = (const float*)d_in[16];
    const float* gate_w   = (const float*)d_in[17];
    const float* w1       = (const float*)d_in[18];
    const float* w2       = (const float*)d_in[19];
    const float* w3       = (const float*)d_in[20];
    const float* w3_b     = (const float*)d_in[21];

    float* out = (float*)d_out;
    float* ws  = (float*)d_ws;

    // workspace layout (floats); later stages reuse earlier regions
    float* h      = ws;                         // 2048*1024   (reused as h2)
    float* qa     = h      + (size_t)R_ * D_;   // 2048*512
    float* q      = qa     + (size_t)R_ * QL_;  // 2048*3072   (reused as o)
    float* kv     = q      + (size_t)R_ * (H_ * QKH_);           // 2048*576
    float* q_ext  = kv     + (size_t)R_ * 576;  // 2048*16*576 (reused as ctx)
    float* k_ext  = q_ext  + (size_t)R_ * H_ * 576;              // 2048*576
    float* scores = k_ext  + (size_t)R_ * 576;  // 32*1024*1024
    float* h1     = scores + (size_t)B_ * H_ * S_ * S_;          // 2048*2048
    float* tbuf   = h1     + (size_t)R_ * DFF_; // 2048*2048
    float* comb   = tbuf   + (size_t)R_ * DFF_; // 2048*8
    float* o      = q;       // alias: q consumed before o is produced
    float* ctx    = q_ext;   // alias: q_ext consumed before ctx is produced
    float* h2     = h;       // alias: h consumed before h2 is produced

    const float SCALE = 0.07216878364870322f;   // 192^-0.5
    dim3 blk(256);

    // 1) h = rms(x, norm1_w)
    rms_kernel<<<R_, blk, 0, stream>>>(x, D_, norm1_w, h, D_, D_);

    // 2) qa = h @ wq_a_w.T + b   [2048 x 512], K=1024
    gemm_wmma_kernel<EP_BIAS, 1><<<dim3(QL_/64, R_/128, 1), blk, 0, stream>>>(
        h, D_, BZ, wq_a_w, D_, BZ, qa, QL_, BZ,
        wq_a_b, BZ, nullptr, 0, BZ, nullptr, 0, BZ, 0.f, R_, QL_, D_);

    // 3) qa = rms(qa, q_norm_w) in place
    rms_kernel<<<R_, blk, 0, stream>>>(qa, QL_, q_norm_w, qa, QL_, QL_);

    // 4) q = qa @ wq_b_w.T + b   [2048 x 3072], K=512
    gemm_wmma_kernel<EP_BIAS, 1><<<dim3((H_*QKH_)/64, R_/128, 1), blk, 0, stream>>>(
        qa, QL_, BZ, wq_b_w, QL_, BZ, q, H_*QKH_, BZ,
        wq_b_b, BZ, nullptr, 0, BZ, nullptr, 0, BZ, 0.f, R_, H_*QKH_, QL_);

    // 5) kv = h @ wkv_a_w.T + b  [2048 x 576], K=1024
    gemm_wmma_kernel<EP_BIAS, 1><<<dim3(576/64, R_/128, 1), blk, 0, stream>>>(
        h, D_, BZ, wkv_a_w, D_, BZ, kv, 576, BZ,
        wkv_a_b, BZ, nullptr, 0, BZ, nullptr, 0, BZ, 0.f, R_, 576, D_);

    // 6) k_ext[:,0:512] = rms(kv[:,0:512], kv_norm_w)   (kv_proj)
    rms_kernel<<<R_, blk, 0, stream>>>(kv, 576, kv_norm_w, k_ext, 576, KVL_);

    // 7) k_ext[:,512:576] = rope(kv[:,512:576])
    rope_k_kernel<<<(R_*32 + 255)/256, blk, 0, stream>>>(kv, fcos, fsin, k_ext);

    // 8) qn: per-head  q_nope[2048x128] @ wkv_b[h,:128,:].T -> q_ext[:,h,0:512]
    {
        BOff ba = {QKH_, 0, 1};                     // A offset = h*192
        BOff bw = {(long)(NOPE_+VH_)*KVL_, 0, 1};   // W offset = h*256*512
        BOff bc = {576, 0, 1};                      // C offset = h*576
        gemm_wmma_kernel<EP_NONE, 1><<<dim3(KVL_/64, R_/128, H_), blk, 0, stream>>>(
            q, H_*QKH_, ba, wkv_b_w, KVL_, bw, q_ext, H_*576, bc,
            nullptr, BZ, nullptr, 0, BZ, nullptr, 0, BZ, 0.f, R_, KVL_, NOPE_);
    }

    // 9) q_ext[:,h,512:576] = rope(q_pe)
    rope_q_kernel<<<(R_*H_*32 + 255)/256, blk, 0, stream>>>(q, fcos, fsin, q_ext);

    // 10) scores[z= b*H+h][s][t] = (q_ext . k_ext)*SCALE + mask[s][t]   K=576
    {
        BOff ba = {(long)S_*H_*576, 576, H_};       // b*S*H*576 + h*576
        BOff bw = {(long)S_*576, 0, H_};            // b*S*576
        BOff bc = {(long)S_*S_, 0, 1};              // z*S*S
        gemm_wmma_kernel<EP_SCALE_MASK, 1><<<dim3(S_/64, S_/128, B_*H_), blk, 0, stream>>>(
            q_ext, H_*576, ba, k_ext, 576, bw, scores, S_, bc,
            nullptr, BZ, mask, S_, BZ, nullptr, 0, BZ, SCALE, S_, S_, 576);
    }

    // 11) softmax rows of scores
    softmax_kernel<<<B_*H_*S_, blk, 0, stream>>>(scores, S_);

    // 12) ctx[z][s][c] = p @ kv_proj   (B not transposed), K=1024, N=512
    {
        BOff ba = {(long)S_*S_, 0, 1};
        BOff bw = {(long)S_*576, 0, H_};
        BOff bc = {(long)S_*H_*KVL_, KVL_, H_};     // b*S*H*512 + h*512
        gemm_wmma_kernel<EP_NONE, 0><<<dim3(KVL_/64, S_/128, B_*H_), blk, 0, stream>>>(
            scores, S_, ba, k_ext, 576, bw, ctx, H_*KVL_, bc,
            nullptr, BZ, nullptr, 0, BZ, nullptr, 0, BZ, 0.f, S_, KVL_, S_);
    }

    // 13) o: per-head ctx[2048x512] @ wkv_b[h,128:,:].T -> o[:, h*128..]
    {
        BOff ba = {KVL_, 0, 1};
        BOff bw = {(long)(NOPE_+VH_)*KVL_, 0, 1};
        BOff bc = {VH_, 0, 1};
        gemm_wmma_kernel<EP_NONE, 1><<<dim3(VH_/64, R_/128, H_), blk, 0, stream>>>(
            ctx, H_*KVL_, ba, wkv_b_w + (size_t)NOPE_*KVL_, KVL_, bw, o, H_*VH_, bc,
            nullptr, BZ, nullptr, 0, BZ, nullptr, 0, BZ, 0.f, R_, VH_, KVL_);
    }

    // 14) out = x + o @ wo_w.T + wo_b   [2048 x 1024], K=2048
    gemm_wmma_kernel<EP_BIAS_RES, 1><<<dim3(D_/64, R_/128, 1), blk, 0, stream>>>(
        o, H_*VH_, BZ, wo_w, H_*VH_, BZ, out, D_, BZ,
        wo_b, BZ, x, D_, BZ, nullptr, 0, BZ, 0.f, R_, D_, H_*VH_);

    // 15) h2 = rms(out, norm2_w)
    rms_kernel<<<R_, blk, 0, stream>>>(out, D_, norm2_w, h2, D_, D_);

    // 16) gate: combine[row][8]
    gate_kernel<<<R_/8, blk, 0, stream>>>(h2, gate_w, comb);

    // 17) MoE (dense over experts, weighted by combine; zero weights = no-op)
    for (int e = 0; e < E_; ++e) {
        const float* w1e = w1 + (size_t)e * DFF_ * D_;
        const float* w3e = w3 + (size_t)e * DFF_ * D_;
        const float* w2e = w2 + (size_t)e * D_ * DFF_;
        const float* b3e = w3_b + (size_t)e * DFF_;
        // h1 = silu(h2 @ w1[e].T)
        gemm_wmma_kernel<EP_SILU, 1><<<dim3(DFF_/64, R_/128, 1), blk, 0, stream>>>(
            h2, D_, BZ, w1e, D_, BZ, h1, DFF_, BZ,
            nullptr, BZ, nullptr, 0, BZ, nullptr, 0, BZ, 0.f, R_, DFF_, D_);
        // t = (h2 @ w3[e].T + w3_b[e]) * h1
        gemm_wmma_kernel<EP_BIAS_MUL, 1><<<dim3(DFF_/64, R_/128, 1), blk, 0, stream>>>(
            h2, D_, BZ, w3e, D_, BZ, tbuf, DFF_, BZ,
            b3e, BZ, h1, DFF_, BZ, nullptr, 0, BZ, 0.f, R_, DFF_, D_);
        // out += combine[:,e] * (t @ w2[e].T)
        gemm_wmma_kernel<EP_ROWSCALE_ACC, 1><<<dim3(D_/64, R_/128, 1), blk, 0, stream>>>(
            tbuf, DFF_, BZ, w2e, DFF_, BZ, out, D_, BZ,
            nullptr, BZ, nullptr, 0, BZ, comb + e, E_, BZ, 0.f, R_, D_, DFF_);
    }
}